// PCFGRNN_9835475107997
// MI455X (gfx1250) — compile-verified
//
#include <hip/hip_runtime.h>
#include <hip/hip_bf16.h>
#include <math.h>

// ---------------------------------------------------------------------------
// PCFG-RNN for MI455X (gfx1250, wave32). All batched matmuls use
// V_WMMA_F32_16X16X4_F32 (full fp32 precision) with double-buffered fragment
// loads. Sequential parts (lex GRU, CKY levels) are persistent-block kernels
// communicating through LDS.
// ---------------------------------------------------------------------------

#define Hh 512
#define NN 64
#define VV 50257

typedef __attribute__((ext_vector_type(2))) float v2f;
typedef __attribute__((ext_vector_type(8))) float v8f;

// ---------------------------------------------------------------------------
// fp32 WMMA GEMM:  C[M,N] = A[M,K] * B + bias(optional, per-N)
//   BT=true : B is weight-layout [N,K], element(k,n) = B[n*ldb + k]
//   BT=false: B is row-major [K,N],   element(k,n) = B[k*ldb + n]
// One wave computes a 16(M) x 64(N) strip: 4 accumulators reuse one A
// fragment -> 4 WMMAs per K-step. K is processed in 16-wide chunks with
// explicit double buffering so fragment loads overlap the matrix pipe.
// OOB rows/cols are address-clamped (their products only feed outputs that
// are never stored). K must be a multiple of 16.
// ---------------------------------------------------------------------------
template <bool BT>
__global__ void gemm_wmma(const float* __restrict__ A, long lda,
                          const float* __restrict__ B, long ldb,
                          const float* __restrict__ bias,
                          float* __restrict__ C, long ldc,
                          int M, int N, int K)
{
    const int lane  = threadIdx.x & 31;
    const int lm    = lane & 15;
    const int half  = lane >> 4;           // lanes 0-15 -> k+0,k+1 ; 16-31 -> k+2,k+3
    const int kOff  = half * 2;
    const int nBase = blockIdx.x * 64;
    const int mTile = blockIdx.y * 16;

    const int mA   = mTile + lm;
    const int mAdr = (mA < M) ? mA : (M - 1);
    const float* Arow = A + (size_t)mAdr * lda;

    // clamped per-tile B column indices / row pointers
    int nAdr[4];
    const float* Brow[4];
#pragma unroll
    for (int t = 0; t < 4; ++t) {
        const int n = nBase + 16 * t + lm;
        nAdr[t] = (n < N) ? n : (N - 1);
        Brow[t] = BT ? (B + (size_t)nAdr[t] * ldb) : B;
    }

    v8f acc0 = {}, acc1 = {}, acc2 = {}, acc3 = {};

    // ping/pong fragment buffers: one 16-wide K chunk = 4 sub-steps
    v2f aP[4], bP[4][4];
    v2f aQ[4], bQ[4][4];

#define LOAD_CHUNK(aB, bB, kc)                                              \
    {                                                                       \
        _Pragma("unroll")                                                   \
        for (int s = 0; s < 4; ++s) {                                       \
            const int ka = (kc) + 4 * s + kOff;                             \
            aB[s] = *(const v2f*)(Arow + ka);                               \
            if (BT) {                                                       \
                _Pragma("unroll")                                           \
                for (int t = 0; t < 4; ++t)                                 \
                    bB[s][t] = *(const v2f*)(Brow[t] + ka);                 \
            } else {                                                        \
                const float* r0 = B + (size_t)ka * ldb;                     \
                const float* r1 = B + (size_t)(ka + 1) * ldb;               \
                _Pragma("unroll")                                           \
                for (int t = 0; t < 4; ++t) {                               \
                    bB[s][t].x = r0[nAdr[t]];                               \
                    bB[s][t].y = r1[nAdr[t]];                               \
                }                                                           \
            }                                                               \
        }                                                                   \
    }

#define COMPUTE_CHUNK(aB, bB)                                               \
    {                                                                       \
        _Pragma("unroll")                                                   \
        for (int s = 0; s < 4; ++s) {                                       \
            acc0 = __builtin_amdgcn_wmma_f32_16x16x4_f32(                   \
                       false, aB[s], false, bB[s][0], (short)0, acc0, false, false); \
            acc1 = __builtin_amdgcn_wmma_f32_16x16x4_f32(                   \
                       false, aB[s], false, bB[s][1], (short)0, acc1, false, false); \
            acc2 = __builtin_amdgcn_wmma_f32_16x16x4_f32(                   \
                       false, aB[s], false, bB[s][2], (short)0, acc2, false, false); \
            acc3 = __builtin_amdgcn_wmma_f32_16x16x4_f32(                   \
                       false, aB[s], false, bB[s][3], (short)0, acc3, false, false); \
        }                                                                   \
    }

    LOAD_CHUNK(aP, bP, 0)
    for (int kc = 0; kc < K; kc += 32) {
        // cacheline prefetch four chunk-pairs ahead
        if (kc + 128 < K) {
            __builtin_prefetch(Arow + kc + 128, 0, 3);
            if (BT) {
#pragma unroll
                for (int t = 0; t < 4; ++t)
                    __builtin_prefetch(Brow[t] + kc + 128, 0, 3);
            } else {
                __builtin_prefetch(B + (size_t)(kc + 128) * ldb + nAdr[0], 0, 3);
            }
        }

        if (kc + 16 < K) LOAD_CHUNK(aQ, bQ, kc + 16)
        COMPUTE_CHUNK(aP, bP)
        if (kc + 32 < K) LOAD_CHUNK(aP, bP, kc + 32)
        if (kc + 16 < K) COMPUTE_CHUNK(aQ, bQ)
    }
#undef LOAD_CHUNK
#undef COMPUTE_CHUNK

    // store (bounds-checked); C/D layout: VGPR r -> row mTile + r + 8*half
    v8f accs[4] = {acc0, acc1, acc2, acc3};
#pragma unroll
    for (int t = 0; t < 4; ++t) {
        const int n = nBase + 16 * t + lm;
        if (n >= N) continue;
        const float bv = bias ? bias[n] : 0.0f;
#pragma unroll
        for (int r = 0; r < 8; ++r) {
            const int m = mTile + r + 8 * half;
            if (m < M)
                C[(size_t)m * ldc + n] = accs[t][r] + bv;
        }
    }
}

// ---------------------------------------------------------------------------
// Embedding gather: embs[t,:] = emb[tokens[t],:]
// ---------------------------------------------------------------------------
__global__ void gather_emb(const int* __restrict__ tokens,
                           const float* __restrict__ emb,
                           float* __restrict__ embs)
{
    const int t   = blockIdx.x;
    const int tok = tokens[t];          // tokens is (N,1) int32
    for (int e = threadIdx.x; e < Hh; e += blockDim.x)
        embs[(size_t)t * Hh + e] = emb[(size_t)tok * Hh + e];
}

__global__ void zero_buf(float* __restrict__ p, int n)
{
    const int i = blockIdx.x * blockDim.x + threadIdx.x;
    if (i < n) p[i] = 0.0f;
}

// ---------------------------------------------------------------------------
// Sequential lex GRU. gi_all = embs@Wihᵀ+bih precomputed by WMMA GEMM.
// One 512-thread block; h lives in LDS; 64 strictly-ordered steps.
// ---------------------------------------------------------------------------
__global__ void lex_gru(const float* __restrict__ gi_all,  // [64,1536]
                        const float* __restrict__ Whh,     // [1536,512]
                        const float* __restrict__ bhh,     // [1536]
                        float* __restrict__ lex_hids)      // [64,512]
{
    __shared__ float h[Hh];
    __shared__ float ghs[3 * Hh];
    const int tid = threadIdx.x;
    h[tid] = 0.0f;
    __syncthreads();

    for (int t = 0; t < NN; ++t) {
#pragma unroll
        for (int c = 0; c < 3; ++c) {
            const int jj = tid + c * Hh;
            const float* wr = Whh + (size_t)jj * Hh;
            float s = bhh[jj];
            for (int u = 0; u < Hh; ++u) s = fmaf(h[u], wr[u], s);
            ghs[jj] = s;
        }
        __syncthreads();

        const float* gi = gi_all + (size_t)t * 3 * Hh;
        const float ir = gi[tid], iz = gi[Hh + tid], ic = gi[2 * Hh + tid];
        const float hr = ghs[tid], hz = ghs[Hh + tid], hc = ghs[2 * Hh + tid];
        const float r  = 1.0f / (1.0f + __expf(-(ir + hr)));
        const float z  = 1.0f / (1.0f + __expf(-(iz + hz)));
        const float cc = tanhf(ic + r * hc);
        const float hn = (1.0f - z) * cc + z * h[tid];
        __syncthreads();
        h[tid] = hn;
        lex_hids[(size_t)t * Hh + tid] = hn;
        __syncthreads();
    }
}

__device__ __forceinline__ float log_sigmoid(float x)
{
    return fminf(x, 0.0f) - log1pf(__expf(-fabsf(x)));
}

// ---------------------------------------------------------------------------
// One CKY level l: one block per span i (spans at the same level are
// independent -> no grid sync). 512 threads = 16 waves.
//   scores[m] = lp[i,i+m] + lp[i+m+1,i+l] + log_sigmoid( G[i,i+m] . h[i+m+1,i+l] )
//   kb = argmax ; hA = GRU(h1b,0) ; hB = GRU(h2b,hA) ; update chart + G row.
// ---------------------------------------------------------------------------
__global__ void __launch_bounds__(512)
cky_level(int l,
          const float* __restrict__ Wc,      // compos_w [512,512]
          const float* __restrict__ sWih,    // [1536,512]
          const float* __restrict__ sWhh,    // [1536,512]
          const float* __restrict__ sbih,    // [1536]
          const float* __restrict__ sbhh,    // [1536]
          float* __restrict__ chart_lp,      // [64,64]
          float* __restrict__ chart_h,       // [64,64,512]
          float* __restrict__ G)             // [64,64,512]
{
    const int i = blockIdx.x;
    const int j = i + l;
    const int tid  = threadIdx.x;
    const int lane = tid & 31;
    const int wave = tid >> 5;

    __shared__ float sc[NN];
    __shared__ float h1[Hh], h2[Hh], hA[Hh], hB[Hh];
    __shared__ float gi[3 * Hh], gh[3 * Hh];
    __shared__ int   kbest;

    // ---- split scores (each wave handles splits m = wave, wave+16, ...) ----
    for (int m = wave; m < l; m += 16) {
        const int k = i + m;
        const float* g1  = G       + ((size_t)i * NN + k) * Hh;
        const float* hh2 = chart_h + ((size_t)(k + 1) * NN + j) * Hh;
        float s = 0.0f;
        for (int t = lane; t < Hh; t += 32) s = fmaf(g1[t], hh2[t], s);
        for (int off = 16; off; off >>= 1) s += __shfl_down(s, off, 32);
        if (lane == 0)
            sc[m] = chart_lp[i * NN + k] + chart_lp[(k + 1) * NN + j] + log_sigmoid(s);
    }
    __syncthreads();

    // ---- first-max argmax (matches jnp.argmax tie-breaking) ----
    if (tid == 0) {
        int   kb   = 0;
        float best = sc[0];
        for (int m = 1; m < l; ++m)
            if (sc[m] > best) { best = sc[m]; kb = m; }
        kbest = kb;
        chart_lp[i * NN + j] = best;
    }
    __syncthreads();
    const int kb = kbest;

    // ---- load best children ----
    {
        const float* h1p = chart_h + ((size_t)i * NN + (i + kb)) * Hh;
        const float* h2p = chart_h + ((size_t)(i + kb + 1) * NN + j) * Hh;
        h1[tid] = h1p[tid];
        h2[tid] = h2p[tid];
    }
    __syncthreads();

    // ---- GRU A: x = h1, h = 0  (gh reduces to bhh) ----
#pragma unroll
    for (int c = 0; c < 3; ++c) {
        const int jj = tid + c * Hh;
        const float* wr = sWih + (size_t)jj * Hh;
        float s = sbih[jj];
        for (int t = 0; t < Hh; ++t) s = fmaf(h1[t], wr[t], s);
        gi[jj] = s;
    }
    __syncthreads();
    {
        const float ir = gi[tid], iz = gi[Hh + tid], ic = gi[2 * Hh + tid];
        const float hr = sbhh[tid], hz = sbhh[Hh + tid], hc = sbhh[2 * Hh + tid];
        const float r  = 1.0f / (1.0f + __expf(-(ir + hr)));
        const float z  = 1.0f / (1.0f + __expf(-(iz + hz)));
        const float cc = tanhf(ic + r * hc);
        hA[tid] = (1.0f - z) * cc;            // + z*0
    }
    __syncthreads();

    // ---- GRU B: x = h2, h = hA ----
#pragma unroll
    for (int c = 0; c < 3; ++c) {
        const int jj = tid + c * Hh;
        const float* wr = sWih + (size_t)jj * Hh;
        const float* wh = sWhh + (size_t)jj * Hh;
        float s  = sbih[jj];
        float s2 = sbhh[jj];
        for (int t = 0; t < Hh; ++t) {
            s  = fmaf(h2[t], wr[t], s);
            s2 = fmaf(hA[t], wh[t], s2);
        }
        gi[jj] = s;
        gh[jj] = s2;
    }
    __syncthreads();
    {
        const float ir = gi[tid], iz = gi[Hh + tid], ic = gi[2 * Hh + tid];
        const float hr = gh[tid], hz = gh[Hh + tid], hc = gh[2 * Hh + tid];
        const float r  = 1.0f / (1.0f + __expf(-(ir + hr)));
        const float z  = 1.0f / (1.0f + __expf(-(iz + hz)));
        const float cc = tanhf(ic + r * hc);
        const float hb = (1.0f - z) * cc + z * hA[tid];
        hB[tid] = hb;
        chart_h[((size_t)i * NN + j) * Hh + tid] = hb;
    }
    __syncthreads();

    // ---- G row for the new entry: G[i,j][g] = sum_h hB[h] * Wc[h,g] ----
    {
        float s = 0.0f;
        for (int t = 0; t < Hh; ++t) s = fmaf(hB[t], Wc[(size_t)t * Hh + tid], s);
        G[((size_t)i * NN + j) * Hh + tid] = s;
    }
}

// ---------------------------------------------------------------------------
// log_softmax along axis 0 (the 64-row axis), in place.
// One thread per vocab column; the 64 rows are L2-resident (<13 MB).
// ---------------------------------------------------------------------------
__global__ void col_logsoftmax(float* __restrict__ logits, int V)
{
    const int c = blockIdx.x * blockDim.x + threadIdx.x;
    if (c >= V) return;
    float mx = -__builtin_inff();
    for (int r = 0; r < NN; ++r) mx = fmaxf(mx, logits[(size_t)r * V + c]);
    float s = 0.0f;
    for (int r = 0; r < NN; ++r) s += __expf(logits[(size_t)r * V + c] - mx);
    const float lse = mx + __logf(s);
    for (int r = 0; r < NN; ++r) logits[(size_t)r * V + c] -= lse;
}

__global__ void row_max_k(const float* __restrict__ lp, int V,
                          float* __restrict__ rmax)
{
    const int r = blockIdx.x;
    float mx = -__builtin_inff();
    for (int c = threadIdx.x; c < V; c += blockDim.x)
        mx = fmaxf(mx, lp[(size_t)r * V + c]);
    __shared__ float red[256];
    red[threadIdx.x] = mx;
    __syncthreads();
    for (int o = 128; o; o >>= 1) {
        if (threadIdx.x < o)
            red[threadIdx.x] = fmaxf(red[threadIdx.x], red[threadIdx.x + o]);
        __syncthreads();
    }
    if (threadIdx.x == 0) rmax[r] = red[0];
}

__global__ void finalize(const float* __restrict__ rmax,
                         const float* __restrict__ chart_lp,
                         float* __restrict__ out, long S)
{
    if (threadIdx.x == 0) {
        float s = 0.0f;
        for (int r = 0; r < NN; ++r) s += rmax[r];
        out[S]     = chart_lp[NN - 1];   // chart_lp[0, n-1]
        out[S + 1] = s;
    }
}

// ---------------------------------------------------------------------------
// Host-side orchestration (all launches on `stream`, graph-capture safe).
// ---------------------------------------------------------------------------
extern "C" void kernel_launch(void* const* d_in, const int* in_sizes, int n_in,
                              void* d_out, int out_size, void* d_ws, size_t ws_size,
                              hipStream_t stream)
{
    (void)in_sizes; (void)n_in; (void)out_size; (void)ws_size;

    const int*   tokens   = (const int*)  d_in[0];
    const float* emb      = (const float*)d_in[1];   // [V,512]
    const float* lex_Wih  = (const float*)d_in[2];   // [1536,512]
    const float* lex_Whh  = (const float*)d_in[3];   // [1536,512]
    const float* lex_bih  = (const float*)d_in[4];
    const float* lex_bhh  = (const float*)d_in[5];
    const float* pos_W    = (const float*)d_in[6];   // [512,512]
    const float* pos_b    = (const float*)d_in[7];
    const float* syn_Wih  = (const float*)d_in[8];
    const float* syn_Whh  = (const float*)d_in[9];
    const float* syn_bih  = (const float*)d_in[10];
    const float* syn_bhh  = (const float*)d_in[11];
    const float* compos_w = (const float*)d_in[12];  // [512,512]

    float* logits = (float*)d_out;                   // [64, V] region of output
    const long S = (long)NN * VV;

    // workspace layout (floats)
    float* ws        = (float*)d_ws;
    float* embs      = ws;                                   // 64*512
    float* lexGi     = embs      + (size_t)NN * Hh;          // 64*1536
    float* lex_hids  = lexGi     + (size_t)NN * 3 * Hh;      // 64*512
    float* chart_lp  = lex_hids  + (size_t)NN * Hh;          // 64*64
    float* chart_h   = chart_lp  + (size_t)NN * NN;          // 64*64*512
    float* Gm        = chart_h   + (size_t)NN * NN * Hh;     // 64*64*512
    float* rmax      = Gm        + (size_t)NN * NN * Hh;     // 64

    const long LDD = (long)(NN + 1) * Hh;   // stride hitting chart diagonal rows

    // 1) embedding gather
    gather_emb<<<NN, 256, 0, stream>>>(tokens, emb, embs);

    // 2) lexGi = embs @ lex_Wihᵀ + lex_bih   (M=64,N=1536,K=512) — WMMA
    gemm_wmma<true><<<dim3((3 * Hh + 63) / 64, NN / 16), 32, 0, stream>>>(
        embs, Hh, lex_Wih, Hh, lex_bih, lexGi, 3 * Hh, NN, 3 * Hh, Hh);

    // 3) sequential lex GRU
    lex_gru<<<1, Hh, 0, stream>>>(lexGi, lex_Whh, lex_bhh, lex_hids);

    // 4) syn_hids straight onto the chart diagonal:
    //    chart_h[i,i,:] = lex_hids @ pos_Wᵀ + pos_b   (ldc = 65*512) — WMMA
    gemm_wmma<true><<<dim3((Hh + 63) / 64, NN / 16), 32, 0, stream>>>(
        lex_hids, Hh, pos_W, Hh, pos_b, chart_h, LDD, NN, Hh, Hh);

    // 5) chart_lp = 0
    zero_buf<<<(NN * NN + 255) / 256, 256, 0, stream>>>(chart_lp, NN * NN);

    // 6) G diagonal: G[i,i,:] = chart_h[i,i,:] @ compos_w — WMMA (B row-major)
    gemm_wmma<false><<<dim3((Hh + 63) / 64, NN / 16), 32, 0, stream>>>(
        chart_h, LDD, compos_w, Hh, nullptr, Gm, LDD, NN, Hh, Hh);

    // 7) logits = lex_hids @ embᵀ  (M=64, N=50257, K=512) — the big WMMA GEMM
    gemm_wmma<true><<<dim3((VV + 63) / 64, NN / 16), 32, 0, stream>>>(
        lex_hids, Hh, emb, Hh, nullptr, logits, VV, NN, VV, Hh);

    // 8) CKY levels (spans within a level are independent -> one block each)
    for (int l = 1; l < NN; ++l)
        cky_level<<<NN - l, Hh, 0, stream>>>(
            l, compos_w, syn_Wih, syn_Whh, syn_bih, syn_bhh,
            chart_lp, chart_h, Gm);

    // 9) log_softmax over axis 0, in place on d_out
    col_logsoftmax<<<(VV + 255) / 256, 256, 0, stream>>>(logits, VV);

    // 10) per-row max, then final scalars
    row_max_k<<<NN, 256, 0, stream>>>(logits, VV, rmax);
    finalize<<<1, 64, 0, stream>>>(rmax, chart_lp, (float*)d_out, S);
}